// SparseFCLayer_63522566308105
// MI455X (gfx1250) — compile-verified
//
#include <hip/hip_runtime.h>
#include <hip/hip_bf16.h>

// ---------------------------------------------------------------------------
// SparseFCLayer on MI455X (gfx1250):
//   W_dense = scatter_add(COO)            (zero + fp32 global atomics)
//   out     = relu(W_dense @ X + bias)    (fp32 WMMA 16x16x4 tiled GEMM,
//                                          LDS tiles fed by the Tensor Data
//                                          Mover when the builtin exists)
// ---------------------------------------------------------------------------

#define N_OUT 4096
#define N_IN  4096
#define BATCH 4096

typedef float v2f __attribute__((ext_vector_type(2)));
typedef float v8f __attribute__((ext_vector_type(8)));
typedef unsigned int u32x4 __attribute__((ext_vector_type(4)));
typedef int i32x4 __attribute__((ext_vector_type(4)));
typedef int i32x8 __attribute__((ext_vector_type(8)));

#if defined(__HIP_DEVICE_COMPILE__) && \
    __has_builtin(__builtin_amdgcn_tensor_load_to_lds) && \
    __has_builtin(__builtin_amdgcn_s_wait_tensorcnt)
#define SFC_HAVE_TDM 1
#else
#define SFC_HAVE_TDM 0
#endif

// ---------------- kernel 1: zero the dense-W workspace ----------------------
__global__ void sfc_zero_kernel(float4* __restrict__ w4, int n4) {
    int i = blockIdx.x * blockDim.x + threadIdx.x;
    if (i < n4) w4[i] = float4{0.f, 0.f, 0.f, 0.f};
}

// ---------------- kernel 2: COO scatter-add into dense W --------------------
__global__ void sfc_scatter_kernel(const float* __restrict__ vals,
                                   const int* __restrict__ rows,
                                   const int* __restrict__ cols,
                                   float* __restrict__ W, int nnz) {
    int i = blockIdx.x * blockDim.x + threadIdx.x;
    if (i < nnz) {
        long long idx = (long long)rows[i] * N_IN + cols[i];
        __hip_atomic_fetch_add(&W[idx], vals[i],
                               __ATOMIC_RELAXED, __HIP_MEMORY_SCOPE_AGENT);
    }
}

// ---------------- TDM helper: 2D strided tile -> LDS (ISA ch.8 D# layout) ---
#if SFC_HAVE_TDM
// LDS byte offset of a __shared__ object (generic -> AS(3) gives LDS offset)
#define SFC_LDS_OFF(p) \
    ((unsigned)(unsigned long long)(__attribute__((address_space(3))) char*)(p))

__device__ __forceinline__ void sfc_tdm_load_2d(
        const void* gptr,            // global address of tile start
        unsigned    lds_off,         // LDS byte offset of tile start
        unsigned    tile_d0,         // contiguous elements per row (dim0)
        unsigned    tile_d1,         // number of rows (dim1)
        unsigned long long row_stride_elts,  // dim0 stride in elements
        unsigned    pad_int_enc,     // pad_interval encoding (4 -> every 32 DW)
        unsigned    pad_amt_enc) {   // pad_amount  encoding (3 -> 4 DW)
    const unsigned long long ga = (unsigned long long)gptr;
    // ---- D# group 0: count=1 | lds_addr | global_addr[56:0] | type=2
    u32x4 g0;
    g0.x = 1u;                                            // count=1 (valid)
    g0.y = lds_off;                                       // lds_addr (bytes)
    g0.z = (unsigned)ga;                                  // global_addr[31:0]
    g0.w = (unsigned)((ga >> 32) & 0x1FFFFFFull)          // global_addr[56:32]
         | (2u << 30);                                    // type = 2 ("image")
    // ---- D# group 1: data_size=4B, pad cfg, tensor/tile dims, dim0 stride
    const unsigned td0 = 1u << 28;   // huge tensor dims: tiles are interior,
    const unsigned td1 = 1u << 28;   // never clip at the OOB check
    i32x8 g1;
    g1[0] = (int)((2u << 16)                 // data_size: 2 -> 4 bytes
                | (1u << 20)                 // pad_enable
                | (pad_int_enc << 22)
                | (pad_amt_enc << 25));      // workgroup_mask = 0 (no cluster)
    g1[1] = (int)((td0 & 0xFFFFu) << 16);    // atomic_barrier=0 | tensor_dim0.lo
    g1[2] = (int)(((td0 >> 16) & 0xFFFFu) | ((td1 & 0xFFFFu) << 16));
    g1[3] = (int)(((td1 >> 16) & 0xFFFFu) | (tile_d0 << 16));   // tile_dim0
    g1[4] = (int)(tile_d1 & 0xFFFFu);        // tile_dim1 | tile_dim2=0 (2D)
    g1[5] = (int)(row_stride_elts & 0xFFFFFFFFull);             // dim0 stride lo
    g1[6] = (int)((row_stride_elts >> 32) & 0xFFFFull);         // hi | dim1_stride=0
    g1[7] = 0;
    // ---- groups 2/3 unused for a 2D tile
    i32x4 gz = {0, 0, 0, 0};
#if __clang_major__ >= 23
    i32x8 gz8 = {0, 0, 0, 0, 0, 0, 0, 0};
    __builtin_amdgcn_tensor_load_to_lds(g0, g1, gz, gz, gz8, 0);
#else
    __builtin_amdgcn_tensor_load_to_lds(g0, g1, gz, gz, 0);
#endif
}
#endif // SFC_HAVE_TDM

// ---------------- kernel 3: tiled fp32 WMMA GEMM + bias + relu --------------
// Block: 256 threads = 8 wave32s, computes a 128x128 tile of out.
// Wave grid 2x4: each wave computes 64x32 via 4x2 tiles of 16x16 accumulators.
// K-loop step 32 (8 sub-steps of the 16x16x4 fp32 WMMA).
#define KT 32
#define AS_STRIDE 36   // 32-float A rows + 4 DW pad  (TDM: interval=4, amount=3)
#define BS_STRIDE 136  // 128-float B rows + 8 DW pad (TDM: interval=6, amount=7)

__launch_bounds__(256, 2)
__global__ void sfc_wmma_gemm_kernel(const float* __restrict__ W,   // [N_OUT, N_IN]
                                     const float* __restrict__ X,   // [N_IN, BATCH]
                                     const float* __restrict__ bias,// [N_OUT]
                                     float* __restrict__ out) {     // [N_OUT, BATCH]
    __shared__ float As[128 * AS_STRIDE];
    __shared__ float Bs[KT * BS_STRIDE];

    const int tid  = threadIdx.x;
    const int lane = tid & 31;
    const int wave = tid >> 5;   // 0..7
    const int wm   = wave >> 2;  // 0..1 : 64 rows of the 128-row tile
    const int wn   = wave & 3;   // 0..3 : 32 cols of the 128-col tile
    const int m0   = blockIdx.y * 128;
    const int n0   = blockIdx.x * 128;

    const int lm = lane & 15;    // row/col within a 16-wide fragment
    const int lh = lane >> 4;    // lane half (ISA 7.12.2 layouts)

    v8f acc[4][2];
#pragma unroll
    for (int mt = 0; mt < 4; ++mt)
#pragma unroll
        for (int nt = 0; nt < 2; ++nt)
            acc[mt][nt] = (v8f)0.f;

#if SFC_HAVE_TDM
    const unsigned ldsA = SFC_LDS_OFF(&As[0]);
    const unsigned ldsB = SFC_LDS_OFF(&Bs[0]);
#endif

    for (int k = 0; k < N_IN; k += KT) {
#if SFC_HAVE_TDM
        // ---- Tensor Data Mover: one wave DMAs both tiles into LDS
        if (wave == 0) {
            // A tile: W[m0..m0+127][k..k+31], row stride N_IN, LDS stride 36
            sfc_tdm_load_2d(&W[(size_t)m0 * N_IN + k], ldsA,
                            KT, 128, (unsigned long long)N_IN, 4u, 3u);
            // B tile: X[k..k+31][n0..n0+127], row stride BATCH, LDS stride 136
            sfc_tdm_load_2d(&X[(size_t)k * BATCH + n0], ldsB,
                            128, KT, (unsigned long long)BATCH, 6u, 7u);
            __builtin_amdgcn_s_wait_tensorcnt(0);
        }
        __syncthreads();
#else
        // ---- fallback: stage tiles through VGPRs
#pragma unroll
        for (int i = 0; i < 4; ++i) {
            int f  = tid + i * 256;        // 1024 float4 chunks of the A tile
            int r  = f >> 3;               // 8 chunks per 32-float row
            int c4 = (f & 7) << 2;
            float4 v = *(const float4*)&W[(size_t)(m0 + r) * N_IN + k + c4];
            *(float4*)&As[r * AS_STRIDE + c4] = v;
        }
#pragma unroll
        for (int i = 0; i < 4; ++i) {
            int f  = tid + i * 256;
            int r  = f >> 5;               // 32 chunks per 128-float row
            int c4 = (f & 31) << 2;
            float4 v = *(const float4*)&X[(size_t)(k + r) * BATCH + n0 + c4];
            *(float4*)&Bs[r * BS_STRIDE + c4] = v;
        }
        __syncthreads();
#endif

#pragma unroll
        for (int ks = 0; ks < KT / 4; ++ks) {
            const int kk = ks * 4;
            // A frag (16x4): lane {m=lm, K=kk+2*lh, kk+2*lh+1}
            v2f afr[4];
#pragma unroll
            for (int mt = 0; mt < 4; ++mt)
                afr[mt] = *(const v2f*)&As[(wm * 64 + mt * 16 + lm) * AS_STRIDE
                                          + kk + lh * 2];
            // B frag (4x16): VGPR v <-> row kk + v + 2*lh, col lm
            v2f bfr[2];
#pragma unroll
            for (int nt = 0; nt < 2; ++nt) {
                const int cb = wn * 32 + nt * 16 + lm;
                bfr[nt].x = Bs[(kk + lh * 2 + 0) * BS_STRIDE + cb];
                bfr[nt].y = Bs[(kk + lh * 2 + 1) * BS_STRIDE + cb];
            }
#pragma unroll
            for (int mt = 0; mt < 4; ++mt)
#pragma unroll
                for (int nt = 0; nt < 2; ++nt)
                    acc[mt][nt] = __builtin_amdgcn_wmma_f32_16x16x4_f32(
                        /*neg_a=*/false, afr[mt],
                        /*neg_b=*/false, bfr[nt],
                        /*c_mod=*/(short)0, acc[mt][nt],
                        /*reuse_a=*/false, /*reuse_b=*/false);
        }
        __syncthreads();
    }

    // ---- epilogue: C/D layout VGPR v <-> row (v + 8*lh), col lm; +bias, relu
#pragma unroll
    for (int mt = 0; mt < 4; ++mt) {
        const int rbase = m0 + wm * 64 + mt * 16 + lh * 8;
#pragma unroll
        for (int nt = 0; nt < 2; ++nt) {
            const int col = n0 + wn * 32 + nt * 16 + lm;
#pragma unroll
            for (int v = 0; v < 8; ++v) {
                const int row = rbase + v;
                float r = acc[mt][nt][v] + bias[row];
                out[(size_t)row * BATCH + col] = fmaxf(r, 0.f);
            }
        }
    }
}

// ---------------------------------------------------------------------------
extern "C" void kernel_launch(void* const* d_in, const int* in_sizes, int n_in,
                              void* d_out, int out_size, void* d_ws, size_t ws_size,
                              hipStream_t stream) {
    const float* X    = (const float*)d_in[0];  // inputs  [N_IN, BATCH]
    const float* vals = (const float*)d_in[1];  // values  [NNZ]
    const float* bias = (const float*)d_in[2];  // biases  [N_OUT]
    const int*   rows = (const int*)d_in[3];    // rows    [NNZ]
    const int*   cols = (const int*)d_in[4];    // cols    [NNZ]
    float*       out  = (float*)d_out;          // [N_OUT, BATCH]
    float*       W    = (float*)d_ws;           // dense W [N_OUT, N_IN] (64 MB)

    // 1) zero dense W
    const int n4 = (N_OUT * N_IN) / 4;
    sfc_zero_kernel<<<(n4 + 255) / 256, 256, 0, stream>>>((float4*)d_ws, n4);

    // 2) scatter-add COO values (duplicates accumulate, matching coalesce)
    const int nnz = in_sizes[1];
    sfc_scatter_kernel<<<(nnz + 255) / 256, 256, 0, stream>>>(vals, rows, cols, W, nnz);

    // 3) WMMA GEMM + bias + relu
    dim3 grid(BATCH / 128, N_OUT / 128);
    sfc_wmma_gemm_kernel<<<grid, 256, 0, stream>>>(W, X, bias, out);
}